// coRNN_62835371540840
// MI455X (gfx1250) — compile-verified
//
#include <hip/hip_runtime.h>
#include <hip/hip_bf16.h>
#include <math.h>

// Problem constants (from reference)
#define T_STEPS 512
#define BATCH   128
#define N_INP   128
#define N_HID   1024
#define N_CH    2
#define N_OUT   10
#define M_TOT   (T_STEPS * BATCH)   // 65536 rows of U
#define N_TOT   (N_CH * N_HID)      // 2048 cols of U

typedef __attribute__((ext_vector_type(2))) float v2f;
typedef __attribute__((ext_vector_type(8))) float v8f;
typedef __attribute__((ext_vector_type(4))) int   v4i;

#define AS1 __attribute__((address_space(1)))
#define AS3 __attribute__((address_space(3)))

#if defined(__has_builtin)
#if __has_builtin(__builtin_amdgcn_global_load_async_to_lds_b128) && \
    __has_builtin(__builtin_amdgcn_s_wait_asynccnt)
#define USE_ASYNC_LDS 1
#endif
#endif

// ---------------------------------------------------------------------------
// Kernel 1: U[M_TOT, 2048] = X[M_TOT, 128] @ Wx^T + Wx_b      (fp32 WMMA)
// Workgroup tile: 128(M) x 64(N), 8 waves (4x2), wave tile 32x32.
// K = 128 staged through LDS in two 64-chunks, stride-68 padding (bank-safe).
// Output staged through LDS for 256B-coalesced float4 stores (512MB write).
// ---------------------------------------------------------------------------
#define TM 128
#define TN 64
#define TK 64
#define LSTR 68   // LDS row stride in floats: lane-m stride = 4 banks, 16B-aligned rows

__global__ __launch_bounds__(256) void corrn_encoder_gemm(
    const float* __restrict__ X,      // [M_TOT, 128]
    const float* __restrict__ W,      // [2048, 128]  (Wx_w, row-major)
    const float* __restrict__ bias,   // [2048]
    float* __restrict__ U)            // [M_TOT, 2048]
{
    __shared__ float lA[TM * LSTR];   // also reused as output staging tile
    __shared__ float lB[TN * LSTR];

    const int m0   = blockIdx.x * TM;
    const int n0   = blockIdx.y * TN;
    const int t    = threadIdx.x;
    const int wave = t >> 5;          // wave32 (gfx1250)
    const int lane = t & 31;
    const int wm   = (wave & 3) * 32; // 4 waves along M
    const int wn   = (wave >> 2) * 32;// 2 waves along N
    const int lm   = lane & 15;
    const int kq   = (lane >> 4) * 2; // half-wave K split per ISA A/B layout

    v8f acc[2][2] = {};               // 2x2 tiles of 16x16 per wave

    for (int kc = 0; kc < N_INP; kc += TK) {
#if USE_ASYNC_LDS
        // CDNA5 async global->LDS staging (no VGPR round-trip), ASYNCcnt tracked.
        #pragma unroll
        for (int j = 0; j < 8; ++j) {
            int linear = j * 1024 + t * 4;
            int row = linear >> 6;
            int col = linear & 63;
            __builtin_amdgcn_global_load_async_to_lds_b128(
                (AS1 v4i*)&X[(size_t)(m0 + row) * N_INP + kc + col],
                (AS3 v4i*)&lA[row * LSTR + col], 0, 0);
        }
        #pragma unroll
        for (int j = 0; j < 4; ++j) {
            int linear = j * 1024 + t * 4;
            int row = linear >> 6;
            int col = linear & 63;
            __builtin_amdgcn_global_load_async_to_lds_b128(
                (AS1 v4i*)&W[(size_t)(n0 + row) * N_INP + kc + col],
                (AS3 v4i*)&lB[row * LSTR + col], 0, 0);
        }
        __builtin_amdgcn_s_wait_asynccnt(0);
#else
        // Stage A chunk: 128 x 64 (8192 floats, 8 x float4 per thread)
        #pragma unroll
        for (int j = 0; j < 8; ++j) {
            int linear = j * 1024 + t * 4;
            int row = linear >> 6;
            int col = linear & 63;
            *(float4*)&lA[row * LSTR + col] =
                *(const float4*)&X[(size_t)(m0 + row) * N_INP + kc + col];
        }
        // Stage B chunk: 64 x 64 (4096 floats, 4 x float4 per thread)
        #pragma unroll
        for (int j = 0; j < 4; ++j) {
            int linear = j * 1024 + t * 4;
            int row = linear >> 6;
            int col = linear & 63;
            *(float4*)&lB[row * LSTR + col] =
                *(const float4*)&W[(size_t)(n0 + row) * N_INP + kc + col];
        }
#endif
        __syncthreads();

        #pragma unroll
        for (int kb = 0; kb < TK; kb += 4) {
            v2f a[2], b[2];
            #pragma unroll
            for (int mi = 0; mi < 2; ++mi) {
                int m = wm + mi * 16 + lm;
                a[mi][0] = lA[m * LSTR + kb + kq];
                a[mi][1] = lA[m * LSTR + kb + kq + 1];
            }
            #pragma unroll
            for (int ni = 0; ni < 2; ++ni) {
                int n = wn + ni * 16 + lm;
                b[ni][0] = lB[n * LSTR + kb + kq];
                b[ni][1] = lB[n * LSTR + kb + kq + 1];
            }
            #pragma unroll
            for (int mi = 0; mi < 2; ++mi)
                #pragma unroll
                for (int ni = 0; ni < 2; ++ni)
                    acc[mi][ni] = __builtin_amdgcn_wmma_f32_16x16x4_f32(
                        false, a[mi], false, b[ni], (short)0, acc[mi][ni],
                        false, false);
        }
        __syncthreads();
    }

    // --- Stage D tiles into LDS (reuse lA: 128 x 64, stride 68) ---
    // ISA C/D layout: VGPR r -> row (lane<16 ? r : r+8), col lane%16.
    const int nn = lane & 15;
    const int mb = (lane >> 4) * 8;
    #pragma unroll
    for (int mi = 0; mi < 2; ++mi)
        #pragma unroll
        for (int ni = 0; ni < 2; ++ni) {
            int trow = wm + mi * 16 + mb;
            int tcol = wn + ni * 16 + nn;
            #pragma unroll
            for (int r = 0; r < 8; ++r)
                lA[(trow + r) * LSTR + tcol] = acc[mi][ni][r];
        }
    __syncthreads();

    // --- Coalesced flush: 16 threads per row, float4, 256B segments ---
    const int fcol = (t & 15) * 4;            // 0..60
    const int frow0 = t >> 4;                 // 0..15
    float4 bv = *(const float4*)&bias[n0 + fcol];
    #pragma unroll
    for (int p = 0; p < 8; ++p) {
        int row = frow0 + p * 16;
        float4 v = *(float4*)&lA[row * LSTR + fcol];
        v.x += bv.x; v.y += bv.y; v.z += bv.z; v.w += bv.w;
        *(float4*)&U[(size_t)(m0 + row) * N_TOT + n0 + fcol] = v;
    }
}

// ---------------------------------------------------------------------------
// Kernel 2: recurrence. One workgroup per batch element (fully independent).
// LC weights/bias register-cached (step-invariant). Hidden state double-
// buffered in LDS => 1 barrier per step. U row software-pipelined through
// registers so global latency never sits on the 512-step serial chain.
// ---------------------------------------------------------------------------
__global__ __launch_bounds__(256) void corrn_recurrence(
    const float* __restrict__ U,      // [T*B, 2048], row (t*B + b)
    const float* __restrict__ Wy_w,   // [2,2,1024,3]
    const float* __restrict__ Wy_b,   // [2,1024]
    const float* __restrict__ R_w,    // [10, 2048]
    const float* __restrict__ R_b,    // [10]
    float* __restrict__ out)          // [128, 10]
{
    __shared__ float h[2][N_TOT];     // double-buffered state, 16 KB
    __shared__ float red[16];

    const int b = blockIdx.x;
    const int t = threadIdx.x;

    // Register-cache step-invariant weights for this thread's 8 (o,j) slots.
    float wreg[4][2][2][3];
    float breg[4][2];
    #pragma unroll
    for (int jj = 0; jj < 4; ++jj) {
        int j = t + jj * 256;
        #pragma unroll
        for (int o = 0; o < 2; ++o) {
            breg[jj][o] = Wy_b[o * N_HID + j];
            #pragma unroll
            for (int c = 0; c < 2; ++c)
                #pragma unroll
                for (int k = 0; k < 3; ++k)
                    wreg[jj][o][c][k] = Wy_w[((o * 2 + c) * N_HID + j) * 3 + k];
        }
    }
    for (int i = t; i < N_TOT; i += 256) h[0][i] = 0.0f;
    if (t < 16) red[t] = 0.0f;
    __syncthreads();

    // Preload u for step 0.
    float ucur[2][4];
    {
        const float* u0 = U + (size_t)b * N_TOT;
        #pragma unroll
        for (int o = 0; o < 2; ++o)
            #pragma unroll
            for (int jj = 0; jj < 4; ++jj)
                ucur[o][jj] = u0[o * N_HID + t + jj * 256];
    }

    int cur = 0;
    for (int step = 0; step < T_STEPS; ++step) {
        // Issue next step's u loads first: latency overlapped with compute.
        float unext[2][4] = {};
        if (step + 1 < T_STEPS) {
            const float* un = U + ((size_t)(step + 1) * BATCH + b) * N_TOT;
            #pragma unroll
            for (int o = 0; o < 2; ++o)
                #pragma unroll
                for (int jj = 0; jj < 4; ++jj)
                    unext[o][jj] = un[o * N_HID + t + jj * 256];
        }

        const float* hc = h[cur];
        float*       hn = h[cur ^ 1];
        #pragma unroll
        for (int jj = 0; jj < 4; ++jj) {
            int j  = t + jj * 256;
            int jm = (j + N_HID - 1) & (N_HID - 1);
            int jp = (j + 1) & (N_HID - 1);
            float h0m = hc[jm],          h00 = hc[j],          h0p = hc[jp];
            float h1m = hc[N_HID + jm],  h10 = hc[N_HID + j],  h1p = hc[N_HID + jp];
            #pragma unroll
            for (int o = 0; o < 2; ++o) {
                float loc = h0m * wreg[jj][o][0][0] + h00 * wreg[jj][o][0][1]
                          + h0p * wreg[jj][o][0][2]
                          + h1m * wreg[jj][o][1][0] + h10 * wreg[jj][o][1][1]
                          + h1p * wreg[jj][o][1][2];
                hn[o * N_HID + j] = tanhf(ucur[o][jj] + loc + breg[jj][o]);
            }
        }
        __syncthreads();   // new state visible; old buffer free to overwrite
        #pragma unroll
        for (int o = 0; o < 2; ++o)
            #pragma unroll
            for (int jj = 0; jj < 4; ++jj)
                ucur[o][jj] = unext[o][jj];
        cur ^= 1;
    }

    // out[b, :] = h_final @ R_w^T + R_b  (fused tail, ds_add_f32 reduction)
    const float* hf = h[cur];
    float partial[N_OUT];
    #pragma unroll
    for (int oo = 0; oo < N_OUT; ++oo) partial[oo] = 0.0f;
    for (int i = t; i < N_TOT; i += 256) {
        float hv = hf[i];
        #pragma unroll
        for (int oo = 0; oo < N_OUT; ++oo)
            partial[oo] += hv * R_w[oo * N_TOT + i];
    }
    #pragma unroll
    for (int oo = 0; oo < N_OUT; ++oo) atomicAdd(&red[oo], partial[oo]);
    __syncthreads();
    if (t < N_OUT) out[b * N_OUT + t] = red[t] + R_b[t];
}

// ---------------------------------------------------------------------------
extern "C" void kernel_launch(void* const* d_in, const int* in_sizes, int n_in,
                              void* d_out, int out_size, void* d_ws, size_t ws_size,
                              hipStream_t stream) {
    const float* x    = (const float*)d_in[0];  // [512,128,128]
    const float* Wx_w = (const float*)d_in[1];  // [2048,128]
    const float* Wx_b = (const float*)d_in[2];  // [2048]
    const float* Wy_w = (const float*)d_in[3];  // [2,2,1024,3]
    const float* Wy_b = (const float*)d_in[4];  // [2,1024]
    const float* R_w  = (const float*)d_in[5];  // [10,2048]
    const float* R_b  = (const float*)d_in[6];  // [10]

    float* U = (float*)d_ws;                    // [65536, 2048] fp32 (512 MB)

    dim3 grid(M_TOT / TM, N_TOT / TN);          // 512 x 32
    corrn_encoder_gemm<<<grid, 256, 0, stream>>>(x, Wx_w, Wx_b, U);
    corrn_recurrence<<<BATCH, 256, 0, stream>>>(U, Wy_w, Wy_b, R_w, R_b,
                                                (float*)d_out);
}